// RelPartialLearnableMultiHeadAttn_28810640622414
// MI455X (gfx1250) — compile-verified
//
#include <hip/hip_runtime.h>
#include <hip/hip_bf16.h>

typedef __attribute__((ext_vector_type(2))) float v2f;
typedef __attribute__((ext_vector_type(8))) float v8f;
typedef unsigned int u32x4 __attribute__((ext_vector_type(4)));
typedef int i32x4 __attribute__((ext_vector_type(4)));
typedef int i32x8 __attribute__((ext_vector_type(8)));

#define QLEN 2048
#define BSZ 2
#define RLEN 2048
#define KLEN 2048
#define NH 16
#define DH 64
#define DM 1024
#define H3 3072              /* 3*NH*DH */
#define ROW3 (BSZ * H3)      /* heads row stride over q = 6144 */
#define ATTN_SCALE 0.125f    /* 1/sqrt(64) */

#define KC 32                            /* k-chunk staged in LDS */
#define PANEL (16 * KC)                  /* floats per A panel */
#define PANEL_BYTES (PANEL * 4)

#if __has_builtin(__builtin_amdgcn_tensor_load_to_lds) && \
    __has_builtin(__builtin_amdgcn_s_wait_tensorcnt)
#define HAVE_TDM 1
#else
#define HAVE_TDM 0
#endif

__device__ __forceinline__ v8f wmma4(v2f a, v2f b, v8f c) {
  // V_WMMA_F32_16X16X4_F32: full-fp32 matrix op (keeps reference precision)
  return __builtin_amdgcn_wmma_f32_16x16x4_f32(false, a, false, b, (short)0, c,
                                               false, false);
}

#if HAVE_TDM
// ---------------------------------------------------------------------------
// TDM: async DMA of a (tile_d1 rows x tile_d0 cols) fp32 tile, row stride
// `stride0` elements, into LDS at byte offset lds_off. D# bitfields per
// CDNA5 ISA (group0: count/lds/global/type,
// group1: data_size + dims/strides). 2-D tensor -> groups 2/3 zero.
// ---------------------------------------------------------------------------
__device__ __forceinline__ void tdm_load_tile(const float* gsrc,
                                              unsigned lds_off, int tensor_d0,
                                              int tensor_d1, int tile_d0,
                                              int tile_d1, long stride0) {
  const unsigned long long ga = (unsigned long long)gsrc;
  u32x4 g0;
  g0.x = 1u;                                   // count=1 (valid user D#)
  g0.y = lds_off;                              // lds_addr (bytes)
  g0.z = (unsigned)(ga & 0xffffffffu);         // global_addr[31:0]
  g0.w = (unsigned)((ga >> 32) & 0x01ffffffu)  // global_addr[56:32]
         | (2u << 30);                         // type = 2 ("image")
  i32x8 g1;
  g1[0] = (int)(2u << 16);                     // data_size=2 (4B); mask=0
  g1[1] = (int)(((unsigned)tensor_d0 & 0xffffu) << 16);
  g1[2] = (int)((((unsigned)tensor_d0 >> 16) & 0xffffu) |
                (((unsigned)tensor_d1 & 0xffffu) << 16));
  g1[3] = (int)((((unsigned)tensor_d1 >> 16) & 0xffffu) |
                (((unsigned)tile_d0 & 0xffffu) << 16));
  g1[4] = (int)((unsigned)tile_d1 & 0xffffu);  // tile_dim2 = 0
  g1[5] = (int)((unsigned long long)stride0 & 0xffffffffu);
  g1[6] = (int)(((unsigned long long)stride0 >> 32) & 0xffffu);
  g1[7] = 0;
  const i32x4 z4 = {0, 0, 0, 0};
#if defined(__clang_major__) && __clang_major__ >= 23
  const i32x8 z8 = {0, 0, 0, 0, 0, 0, 0, 0};
  __builtin_amdgcn_tensor_load_to_lds(g0, g1, z4, z4, z8, 0);
#else
  __builtin_amdgcn_tensor_load_to_lds(g0, g1, z4, z4, 0);
#endif
}

// Explicit DS read of a float2 at LDS byte offset `off`. The TDM populates
// LDS invisibly to the optimizer (the destination is descriptor *data*), so
// a normal load gets folded to undef (round 2) and a volatile generic load
// lowers to flat_load@SYS (round 3). Inline ds_load_b64 is the right path.
__device__ __forceinline__ v2f lds_read_v2f(unsigned off) {
  v2f a;
  asm volatile("ds_load_b64 %0, %1\n\ts_wait_dscnt 0x0"
               : "=v"(a)
               : "v"(off));
  return a;
}
#endif  // HAVE_TDM

// ---------------------------------------------------------------------------
// NN GEMM: C[M,N] = A[M,K] @ B[K,N]. Block = 8 waves, all sharing one 16-row
// A stripe; wave w owns a 16x64 output block (4 accumulators, A-frag reuse).
// A panel (16 x KC) staged in LDS: TDM double-buffered async DMA when
// available, cooperative loads otherwise. Requires N%512==0, K%KC==0.
// ---------------------------------------------------------------------------
__global__ __launch_bounds__(256) void k_gemm_nn(
    const float* __restrict__ A, int lda, const float* __restrict__ B, int ldb,
    float* __restrict__ C, int ldc, int M, int N, int K) {
  __shared__ float sA[2][PANEL];
  const int lane = threadIdx.x & 31;
  const int w = threadIdx.x >> 5;
  const int nb = N >> 9;  // 512-wide column blocks
  const int bi = blockIdx.x / nb, bj = blockIdx.x - bi * nb;
  const int i0 = bi << 4;
  const int jbase = (bj << 9) + (w << 6);
  const int kk = (lane >> 4) << 1;

  int col[4];
#pragma unroll
  for (int t = 0; t < 4; ++t) col[t] = jbase + (t << 4) + (lane & 15);

  v8f acc[4] = {};

#if HAVE_TDM
  // True LDS byte offset of sA: generic pointers to __shared__ carry the
  // AS(3) offset in their low 32 bits (ISA 10.2: LDS aperture uses addr[31:0]).
  const unsigned sbase = (unsigned)(unsigned long long)&sA[0][0];
  const unsigned rowoff = sbase + (unsigned)(((lane & 15) * KC + kk) * 4);
  if (threadIdx.x < 32)
    tdm_load_tile(A + (size_t)i0 * lda, sbase, K, M, KC, 16, lda);
#endif

  for (int kc = 0, it = 0; kc < K; kc += KC, ++it) {
#if HAVE_TDM
    const int buf = it & 1;
    if (threadIdx.x < 32) __builtin_amdgcn_s_wait_tensorcnt((short)0);
    __syncthreads();  // panel `buf` ready for all waves
    if (kc + KC < K && threadIdx.x < 32)
      tdm_load_tile(A + (size_t)i0 * lda + kc + KC,
                    sbase + (unsigned)((buf ^ 1) * PANEL_BYTES), K, M, KC, 16,
                    lda);
    const unsigned bufoff = rowoff + (unsigned)(buf * PANEL_BYTES);
#else
    __syncthreads();  // all waves done with previous panel
    {
      const int row = threadIdx.x >> 4;
      const int cp = (threadIdx.x & 15) << 1;
      *(v2f*)&sA[0][row * KC + cp] =
          *(const v2f*)&A[(size_t)(i0 + row) * lda + kc + cp];
    }
    __syncthreads();
    const float* sa = &sA[0][(lane & 15) * KC];
#endif
    if (kc + KC < K)  // prime next B chunk (global_prefetch_b8)
      __builtin_prefetch(&B[(size_t)(kc + KC) * ldb + col[0]], 0, 1);

#pragma unroll
    for (int k0 = 0; k0 < KC; k0 += 4) {
#if HAVE_TDM
      const v2f a = lds_read_v2f(bufoff + (unsigned)(k0 * 4));  // ds_load_b64
#else
      v2f a;
      a.x = sa[k0 + kk];
      a.y = sa[k0 + kk + 1];
#endif
      const float* Bk0 = B + (size_t)(kc + k0 + kk) * ldb;
      const float* Bk1 = Bk0 + ldb;
#pragma unroll
      for (int t = 0; t < 4; ++t) {
        v2f bb;
        bb.x = Bk0[col[t]];
        bb.y = Bk1[col[t]];
        acc[t] = wmma4(a, bb, acc[t]);
      }
    }
  }

  const int rb = (lane >> 4) * 8;
#pragma unroll
  for (int t = 0; t < 4; ++t)
#pragma unroll
    for (int v = 0; v < 8; ++v)
      C[(size_t)(i0 + rb + v) * ldc + col[t]] = acc[t][v];
}

// ---------------------------------------------------------------------------
// Raw BD GEMM for one (b,n): BD[q,r] = (Wq[q,b,n,:]+bias[n,:]) . Rk[r,n,:]
// NT gemm, K = 64. Wave computes 16x64 (4 tiles, A-frag reuse).
// ---------------------------------------------------------------------------
__global__ __launch_bounds__(256) void k_bd_gemm(
    const float* __restrict__ heads, const float* __restrict__ rheadk,
    const float* __restrict__ bias, float* __restrict__ bd, int b, int n) {
  const int lane = threadIdx.x & 31;
  const int wave = blockIdx.x * (blockDim.x >> 5) + (threadIdx.x >> 5);
  const int ti = wave >> 5, tj = wave & 31;  // 128 x 32 (of 16x64)
  const int i0 = ti << 4, j0 = tj << 6;
  const int m = i0 + (lane & 15);
  const int kk = (lane >> 4) << 1;

  const float* Arow = heads + (size_t)m * ROW3 + (size_t)b * H3 + n * DH;
  const float* bi = bias + n * DH;
  const float* Brow[4];
  int col[4];
#pragma unroll
  for (int t = 0; t < 4; ++t) {
    col[t] = j0 + (t << 4) + (lane & 15);
    Brow[t] = rheadk + (size_t)col[t] * DM + n * DH;
  }

  v8f acc[4] = {};
#pragma unroll
  for (int k0 = 0; k0 < DH; k0 += 4) {
    v2f a;
    a.x = Arow[k0 + kk] + bi[k0 + kk];
    a.y = Arow[k0 + kk + 1] + bi[k0 + kk + 1];
#pragma unroll
    for (int t = 0; t < 4; ++t) {
      v2f bb;
      bb.x = Brow[t][k0 + kk];  // NT: contiguous along K
      bb.y = Brow[t][k0 + kk + 1];
      acc[t] = wmma4(a, bb, acc[t]);
    }
  }
  const int rb = (lane >> 4) * 8;
#pragma unroll
  for (int t = 0; t < 4; ++t)
#pragma unroll
    for (int v = 0; v < 8; ++v)
      bd[(size_t)(i0 + rb + v) * RLEN + col[t]] = acc[t][v];
}

// ---------------------------------------------------------------------------
// Scores for one (b,n): AC via WMMA (NT, K=64, 16x64 per wave) + rel-shifted
// BD gather + scale + key-padding mask -> coverage region of d_out.
// ---------------------------------------------------------------------------
__global__ __launch_bounds__(256) void k_score(
    const float* __restrict__ heads, const float* __restrict__ bd,
    const float* __restrict__ bias, const unsigned char* __restrict__ mask,
    float* __restrict__ cov, int b, int n) {
  const int lane = threadIdx.x & 31;
  const int wave = blockIdx.x * (blockDim.x >> 5) + (threadIdx.x >> 5);
  const int ti = wave >> 5, tj = wave & 31;
  const int i0 = ti << 4, j0 = tj << 6;
  const int m = i0 + (lane & 15);
  const int kk = (lane >> 4) << 1;

  const float* Arow = heads + (size_t)m * ROW3 + (size_t)b * H3 + n * DH;
  const float* bi = bias + n * DH;
  const float* Brow[4];
  int col[4];
  bool mk[4];
#pragma unroll
  for (int t = 0; t < 4; ++t) {
    col[t] = j0 + (t << 4) + (lane & 15);
    Brow[t] = heads + (size_t)col[t] * ROW3 + (size_t)b * H3 + DM + n * DH;
    mk[t] = mask[col[t] * BSZ + b] != 0;
  }

  v8f acc[4] = {};
#pragma unroll
  for (int k0 = 0; k0 < DH; k0 += 4) {
    v2f a;
    a.x = Arow[k0 + kk] + bi[k0 + kk];
    a.y = Arow[k0 + kk + 1] + bi[k0 + kk + 1];
#pragma unroll
    for (int t = 0; t < 4; ++t) {
      v2f bb;
      bb.x = Brow[t][k0 + kk];
      bb.y = Brow[t][k0 + kk + 1];
      acc[t] = wmma4(a, bb, acc[t]);
    }
  }

  const int rb = (lane >> 4) * 8;
  float* crow = cov + (size_t)(b * NH + n) * QLEN * KLEN;
#pragma unroll
  for (int t = 0; t < 4; ++t) {
    const int j = col[t];
#pragma unroll
    for (int v = 0; v < 8; ++v) {
      const int i = i0 + rb + v;
      const int d = j - i;
      float bdv;
      if (d == 1)
        bdv = 0.0f;
      else if (d <= 0)
        bdv = bd[(size_t)i * RLEN + (RLEN - 1 + d)];
      else
        bdv = bd[(size_t)(i + 1) * RLEN + (d - 2)];
      float s = (acc[t][v] + bdv) * ATTN_SCALE;
      if (mk[t]) s = -__builtin_inff();
      crow[(size_t)i * KLEN + j] = s;
    }
  }
}

// ---------------------------------------------------------------------------
// Row-wise softmax over coverage (in place); all -inf row -> zeros (NaN fix).
// ---------------------------------------------------------------------------
__global__ __launch_bounds__(256) void k_softmax(float* __restrict__ cov) {
  __shared__ float red[256];
  float* p = cov + (size_t)blockIdx.x * KLEN;
  const int t = threadIdx.x;

  float m = -__builtin_inff();
  for (int j = t; j < KLEN; j += 256) m = fmaxf(m, p[j]);
  red[t] = m;
  __syncthreads();
  for (int s = 128; s > 0; s >>= 1) {
    if (t < s) red[t] = fmaxf(red[t], red[t + s]);
    __syncthreads();
  }
  m = red[0];
  __syncthreads();

  if (m == -__builtin_inff()) {  // fully masked row -> softmax NaN -> 0
    for (int j = t; j < KLEN; j += 256) p[j] = 0.0f;
    return;
  }
  float sum = 0.0f;
  for (int j = t; j < KLEN; j += 256) {
    const float e = expf(p[j] - m);
    p[j] = e;
    sum += e;
  }
  red[t] = sum;
  __syncthreads();
  for (int s = 128; s > 0; s >>= 1) {
    if (t < s) red[t] += red[t + s];
    __syncthreads();
  }
  const float inv = 1.0f / red[0];
  for (int j = t; j < KLEN; j += 256) p[j] *= inv;
}

// ---------------------------------------------------------------------------
// attn_vec[q,b,n*64+dh] = sum_k P[b,n,q,k] * Wv[k,b,n,dh]; K=2048.
// One (b,n) per blockIdx.z; wave computes all 64 cols for 16 rows
// (contiguous P rows -> global_load_b64 for A; 4 WMMAs reuse the A frag).
// ---------------------------------------------------------------------------
__global__ __launch_bounds__(256) void k_pv(const float* __restrict__ cov,
                                            const float* __restrict__ heads,
                                            float* __restrict__ attn_vec) {
  const int z = blockIdx.z;  // b*16 + n
  const int b = z >> 4, n = z & 15;
  const int lane = threadIdx.x & 31;
  const int ti = blockIdx.x * (blockDim.x >> 5) + (threadIdx.x >> 5);
  const int i0 = ti << 4;
  const int m = i0 + (lane & 15);
  const int kk = (lane >> 4) << 1;

  const float* P = cov + (size_t)z * QLEN * KLEN + (size_t)m * KLEN;
  const float* Vc[4];
#pragma unroll
  for (int t = 0; t < 4; ++t)
    Vc[t] = heads + (size_t)b * H3 + 2 * DM + n * DH + (t << 4) + (lane & 15);

  v8f acc[4] = {};
#pragma unroll 2
  for (int k0 = 0; k0 < KLEN; k0 += 4) {
    const v2f a = *(const v2f*)(P + k0 + kk);  // contiguous prob row
#pragma unroll
    for (int t = 0; t < 4; ++t) {
      v2f bb;
      bb.x = Vc[t][(size_t)(k0 + kk) * ROW3];
      bb.y = Vc[t][(size_t)(k0 + kk + 1) * ROW3];
      acc[t] = wmma4(a, bb, acc[t]);
    }
  }
  const int rb = (lane >> 4) * 8;
  float* out = attn_vec + (size_t)b * DM + n * DH;
#pragma unroll
  for (int t = 0; t < 4; ++t)
#pragma unroll
    for (int v = 0; v < 8; ++v)
      out[(size_t)(i0 + rb + v) * (BSZ * DM) + (t << 4) + (lane & 15)] =
          acc[t][v];
}

// ---------------------------------------------------------------------------
extern "C" void kernel_launch(void* const* d_in, const int* in_sizes, int n_in,
                              void* d_out, int out_size, void* d_ws,
                              size_t ws_size, hipStream_t stream) {
  (void)in_sizes; (void)n_in; (void)out_size; (void)ws_size;
  const float* w = (const float*)d_in[0];        // (2048,2,1024)
  const float* r = (const float*)d_in[1];        // (2048,1024)
  const unsigned char* mask = (const unsigned char*)d_in[2];  // (2048,2) bool
  const float* qkv_w = (const float*)d_in[3];    // (1024,3072)
  const float* r_net_w = (const float*)d_in[4];  // (1024,1024)
  const float* o_w = (const float*)d_in[5];      // (1024,1024)
  // d_in[6] = r_w_bias: unused (reference bug uses r_r_bias for both)
  const float* r_r_bias = (const float*)d_in[7]; // (16,64)

  float* out = (float*)d_out;                  // (2048,2,1024)
  float* cov = out + (size_t)QLEN * BSZ * DM;  // (2,16,2048,2048)

  float* heads = (float*)d_ws;                      // 50.3 MB
  float* rheadk = heads + (size_t)QLEN * BSZ * H3;  // 8.4 MB
  float* bdslab = rheadk + (size_t)RLEN * DM;       // 16.8 MB
  float* attn_vec = bdslab + (size_t)QLEN * RLEN;   // 16.8 MB

  // 1) heads = w @ qkv_w : M=4096 N=3072 K=1024
  k_gemm_nn<<<(QLEN * BSZ / 16) * (H3 / 512), 256, 0, stream>>>(
      w, DM, qkv_w, H3, heads, H3, QLEN * BSZ, H3, DM);
  // 2) r_head_k = r @ r_net_w : M=2048 N=1024 K=1024
  k_gemm_nn<<<(RLEN / 16) * (DM / 512), 256, 0, stream>>>(
      r, DM, r_net_w, DM, rheadk, DM, RLEN, DM, DM);
  // 3) per (b,n): raw BD GEMM into slab, then fused AC + rel-shift + mask
  //    scores into the coverage region (stream order keeps the slab safe).
  for (int b = 0; b < BSZ; ++b)
    for (int n = 0; n < NH; ++n) {
      k_bd_gemm<<<512, 256, 0, stream>>>(heads, rheadk, r_r_bias, bdslab, b, n);
      k_score<<<512, 256, 0, stream>>>(heads, bdslab, r_r_bias, mask, cov, b, n);
    }
  // 4) softmax over every (b,n,q) row of coverage, in place
  k_softmax<<<BSZ * NH * QLEN, 256, 0, stream>>>(cov);
  // 5) attn_vec = P @ V
  k_pv<<<dim3(16, 1, BSZ * NH), 256, 0, stream>>>(cov, heads, attn_vec);
  // 6) output = attn_vec @ o_w : M=4096 N=1024 K=1024
  k_gemm_nn<<<(QLEN * BSZ / 16) * (DM / 512), 256, 0, stream>>>(
      attn_vec, DM, o_w, DM, out, DM, QLEN * BSZ, DM, NH * DH);
}